// GCN_24395414241686
// MI455X (gfx1250) — compile-verified
//
#include <hip/hip_runtime.h>
#include <math.h>

typedef __attribute__((ext_vector_type(2))) float v2f;
typedef __attribute__((ext_vector_type(8))) float v8f;

// ---------------------------------------------------------------------------
// Degree / normalization kernels
// ---------------------------------------------------------------------------
__global__ void deg_init_kernel(float* __restrict__ deg, int N) {
    int i = blockIdx.x * blockDim.x + threadIdx.x;
    if (i < N) deg[i] = 1.0f;  // self-loop weight
}

__global__ void deg_scatter_kernel(const long long* __restrict__ ei,
                                   const float* __restrict__ w,
                                   float* __restrict__ deg, int E) {
    int e = blockIdx.x * blockDim.x + threadIdx.x;
    if (e < E) {
        long long d = ei[(long long)E + e];  // row 1 = dst
        atomicAdd(&deg[d], w[e]);
    }
}

__global__ void deg_to_dis_kernel(float* __restrict__ deg, int N) {
    int i = blockIdx.x * blockDim.x + threadIdx.x;
    if (i < N) {
        float d = deg[i];
        deg[i] = (d > 0.0f) ? rsqrtf(fmaxf(d, 1e-12f)) : 0.0f;
    }
}

// ---------------------------------------------------------------------------
// Dense GEMM with V_WMMA_F32_16X16X4_F32  (C[M,N] = A[M,K] @ B[K,N], fp32)
// One wave computes a 16 x (NT*16) output tile. M must be a multiple of 16.
//
// f32 WMMA lane layouts (ISA 7.12.2):
//   A 16x4 : lane(l&15)=row M; half=l>>4; VGPR0=K(k+2*half), VGPR1=K(k+1+2*half)
//   B 4x16 : lane(l&15)=col N; half=l>>4; VGPR0=K(k+2*half), VGPR1=K(k+1+2*half)
//   C 16x16: VGPR r, lane l -> M = r + 8*(l>>4), N = l&15
// ---------------------------------------------------------------------------
template <int NT>
__global__ void gemm_wmma_f32_kernel(const float* __restrict__ A,
                                     const float* __restrict__ B,
                                     float* __restrict__ C,
                                     int M, int K) {
    const int N = NT * 16;
    const int lane = threadIdx.x & 31;
    const int wave = threadIdx.x >> 5;
    const int gwave = blockIdx.x * (blockDim.x >> 5) + wave;
    const int m0 = gwave * 16;
    if (m0 >= M) return;  // wave-uniform: EXEC stays all-ones inside

    const int half = lane >> 4;   // 0: K pair {k,k+1}; 1: K pair {k+2,k+3}
    const int l16  = lane & 15;

    v8f acc[NT];
#pragma unroll
    for (int t = 0; t < NT; ++t) acc[t] = (v8f){0,0,0,0,0,0,0,0};

    const float* arow = A + (size_t)(m0 + l16) * K;

    for (int k = 0; k < K; k += 4) {
        v2f a;
        a.x = arow[k + 2 * half];
        a.y = arow[k + 2 * half + 1];
#pragma unroll
        for (int t = 0; t < NT; ++t) {
            v2f b;
            b.x = B[(size_t)(k + 2 * half)     * N + t * 16 + l16];
            b.y = B[(size_t)(k + 2 * half + 1) * N + t * 16 + l16];
            acc[t] = __builtin_amdgcn_wmma_f32_16x16x4_f32(
                /*neg_a=*/false, a, /*neg_b=*/false, b,
                /*c_mod=*/(short)0, acc[t], /*reuse_a=*/false, /*reuse_b=*/false);
        }
    }

#pragma unroll
    for (int t = 0; t < NT; ++t) {
#pragma unroll
        for (int r = 0; r < 8; ++r) {
            int m = m0 + r + 8 * half;
            C[(size_t)m * N + t * 16 + l16] = acc[t][r];
        }
    }
}

// ---------------------------------------------------------------------------
// Aggregation: agg[i,:] = dis[i]^2 * h[i,:]  (self-loop init, also clears poison)
// ---------------------------------------------------------------------------
__global__ void self_init_kernel(const float* __restrict__ h,
                                 const float* __restrict__ dis,
                                 float* __restrict__ agg, int N, int F) {
    long long t = (long long)blockIdx.x * blockDim.x + threadIdx.x;
    long long total = (long long)N * F;
    if (t >= total) return;
    int i = (int)(t / F);
    float d = dis[i];
    agg[t] = h[t] * d * d;
}

// ---------------------------------------------------------------------------
// Edge scatter: agg[dst,:] += dis[src]*w*dis[dst] * h[src,:]
// F/4 lanes per edge, float4 gather + 4 global fp32 atomics each.
// ---------------------------------------------------------------------------
__global__ void edge_scatter_kernel(const float* __restrict__ h,
                                    const long long* __restrict__ ei,
                                    const float* __restrict__ w,
                                    const float* __restrict__ dis,
                                    float* __restrict__ agg,
                                    int E, int F) {
    const int gpe = F >> 2;  // float4 groups per edge
    long long t = (long long)blockIdx.x * blockDim.x + threadIdx.x;
    long long total = (long long)E * gpe;
    if (t >= total) return;
    int e = (int)(t / gpe);
    int g = (int)(t % gpe);
    long long s = ei[e];
    long long d = ei[(long long)E + e];
    float norm = dis[s] * w[e] * dis[d];
    float4 v = *(const float4*)(h + (size_t)s * F + 4 * g);
    float* ap = agg + (size_t)d * F + 4 * g;
    atomicAdd(ap + 0, v.x * norm);
    atomicAdd(ap + 1, v.y * norm);
    atomicAdd(ap + 2, v.z * norm);
    atomicAdd(ap + 3, v.w * norm);
}

// ---------------------------------------------------------------------------
// out[t] = relu(agg[t] + bias[t % F])
// ---------------------------------------------------------------------------
__global__ void bias_relu_kernel(const float* __restrict__ agg,
                                 const float* __restrict__ bias,
                                 float* __restrict__ out, int N, int F) {
    long long t = (long long)blockIdx.x * blockDim.x + threadIdx.x;
    long long total = (long long)N * F;
    if (t >= total) return;
    float v = agg[t] + bias[(int)(t % F)];
    out[t] = v > 0.0f ? v : 0.0f;
}

// ---------------------------------------------------------------------------
// Fused head: out[i] = sigmoid( sum_f (agg2[i,f]+b2[f]) * Wl[f] + bl )
// ---------------------------------------------------------------------------
__global__ void head_kernel(const float* __restrict__ agg2,
                            const float* __restrict__ b2,
                            const float* __restrict__ Wl,
                            const float* __restrict__ bl,
                            float* __restrict__ out, int N, int F) {
    int i = blockIdx.x * blockDim.x + threadIdx.x;
    if (i >= N) return;
    float z = bl[0];
    const float* row = agg2 + (size_t)i * F;
#pragma unroll 8
    for (int f = 0; f < F; ++f) z += (row[f] + b2[f]) * Wl[f];
    out[i] = 1.0f / (1.0f + expf(-z));
}

// ---------------------------------------------------------------------------
extern "C" void kernel_launch(void* const* d_in, const int* in_sizes, int n_in,
                              void* d_out, int out_size, void* d_ws, size_t ws_size,
                              hipStream_t stream) {
    const float*      x  = (const float*)d_in[0];       // [N,128]
    const long long*  ei = (const long long*)d_in[1];   // [2,E] int64
    const float*      ew = (const float*)d_in[2];       // [E]
    const float*      W1 = (const float*)d_in[3];       // [128,64]
    const float*      b1 = (const float*)d_in[4];       // [64]
    const float*      W2 = (const float*)d_in[5];       // [64,32]
    const float*      b2 = (const float*)d_in[6];       // [32]
    const float*      Wl = (const float*)d_in[7];       // [32,1]
    const float*      bl = (const float*)d_in[8];       // [1]
    float* out = (float*)d_out;

    const int IN = 128, H2 = 64, H = 32;
    const int N = in_sizes[0] / IN;
    const int E = in_sizes[2];

    // Workspace layout (floats)
    float* ws = (float*)d_ws;
    size_t o = 0;
    float* dis  = ws + o; o += ((size_t)N + 3) & ~(size_t)3;
    float* h    = ws + o; o += (size_t)N * H2;   // h1 pre/post activation
    float* agg1 = ws + o; o += (size_t)N * H2;
    float* h2   = ws + o; o += (size_t)N * H;
    float* agg2 = ws + o; o += (size_t)N * H;

    const int B = 256;
    auto blocks = [](long long total, int b) { return (int)((total + b - 1) / b); };

    // --- normalization: dis = rsqrt(deg), deg = 1 + sum_in w ---
    deg_init_kernel<<<blocks(N, B), B, 0, stream>>>(dis, N);
    deg_scatter_kernel<<<blocks(E, B), B, 0, stream>>>(ei, ew, dis, E);
    deg_to_dis_kernel<<<blocks(N, B), B, 0, stream>>>(dis, N);

    // --- layer 1: h = x @ W1 (WMMA), aggregate, +b1, relu ---
    {
        int waves = N / 16;                      // N divisible by 16 (50000)
        int grid  = (waves + 7) / 8;             // 8 waves / 256-thread block
        gemm_wmma_f32_kernel<4><<<grid, B, 0, stream>>>(x, W1, h, N, IN);
    }
    self_init_kernel<<<blocks((long long)N * H2, B), B, 0, stream>>>(h, dis, agg1, N, H2);
    edge_scatter_kernel<<<blocks((long long)E * (H2 / 4), B), B, 0, stream>>>(
        h, ei, ew, dis, agg1, E, H2);
    bias_relu_kernel<<<blocks((long long)N * H2, B), B, 0, stream>>>(agg1, b1, h, N, H2);

    // --- layer 2: h2 = h1 @ W2 (WMMA), aggregate ---
    {
        int waves = N / 16;
        int grid  = (waves + 7) / 8;
        gemm_wmma_f32_kernel<2><<<grid, B, 0, stream>>>(h, W2, h2, N, H2);
    }
    self_init_kernel<<<blocks((long long)N * H, B), B, 0, stream>>>(h2, dis, agg2, N, H);
    edge_scatter_kernel<<<blocks((long long)E * (H / 4), B), B, 0, stream>>>(
        h2, ei, ew, dis, agg2, E, H);

    // --- head: sigmoid((agg2 + b2) @ Wl + bl) ---
    head_kernel<<<blocks(N, B), B, 0, stream>>>(agg2, b2, Wl, bl, out, N, H);
}